// Raymarcher_32375463477274
// MI455X (gfx1250) — compile-verified
//
#include <hip/hip_runtime.h>
#include <math.h>
#include <stdint.h>

// Problem constants (match reference)
#define G     128
#define GG    (G * G)
#define GGG   (G * G * G)
#define HH    512
#define WW    512
#define NRAYS (HH * WW)
#define DTC   0.02f   // DT / VOLRADIUS

__device__ __forceinline__ float4 f4zero() { return make_float4(0.f, 0.f, 0.f, 0.f); }

__device__ __forceinline__ float4 lerp4(float4 a, float4 b, float t) {
    return make_float4(fmaf(t, b.x - a.x, a.x),
                       fmaf(t, b.y - a.y, a.y),
                       fmaf(t, b.z - a.z, a.z),
                       fmaf(t, b.w - a.w, a.w));
}

// ---------------------------------------------------------------------------
// Trilinear sample, grid_sample semantics (align_corners=True, zeros padding).
// Interleaved layout: vol[z*GG + y*G + x] is a float4 voxel -> global_load_b128
// via SADDR(base) + 32-bit voffset*16 (scale_offset): flat int indices only,
// no per-corner 64-bit address chains.
// ---------------------------------------------------------------------------
__device__ __forceinline__ float4 tri_sample_i(const float4* __restrict__ vol,
                                               float px, float py, float pz) {
    const float s = 0.5f * (float)(G - 1);   // 63.5
    float fx = fmaf(px, s, s);
    float fy = fmaf(py, s, s);
    float fz = fmaf(pz, s, s);
    float x0f = floorf(fx), y0f = floorf(fy), z0f = floorf(fz);
    float wx = fx - x0f, wy = fy - y0f, wz = fz - z0f;
    int x0 = (int)x0f, y0 = (int)y0f, z0 = (int)z0f;
    int x1 = x0 + 1,  y1 = y0 + 1,  z1 = z0 + 1;
    bool bx0 = (unsigned)x0 < (unsigned)G, bx1 = (unsigned)x1 < (unsigned)G;
    bool by0 = (unsigned)y0 < (unsigned)G, by1 = (unsigned)y1 < (unsigned)G;
    bool bz0 = (unsigned)z0 < (unsigned)G, bz1 = (unsigned)z1 < (unsigned)G;
    int cx0 = min(max(x0, 0), G - 1), cx1 = min(max(x1, 0), G - 1);
    int cy0 = min(max(y0, 0), G - 1), cy1 = min(max(y1, 0), G - 1);
    int cz0 = min(max(z0, 0), G - 1), cz1 = min(max(z1, 0), G - 1);

    int r00 = cz0 * GG + cy0 * G;
    int r01 = cz0 * GG + cy1 * G;
    int r10 = cz1 * GG + cy0 * G;
    int r11 = cz1 * GG + cy1 * G;

    // Zeroed OOB corners make the lerp chain identical to zeros-padded sum.
    float4 v000 = (bz0 && by0 && bx0) ? vol[r00 + cx0] : f4zero();
    float4 v001 = (bz0 && by0 && bx1) ? vol[r00 + cx1] : f4zero();
    float4 v010 = (bz0 && by1 && bx0) ? vol[r01 + cx0] : f4zero();
    float4 v011 = (bz0 && by1 && bx1) ? vol[r01 + cx1] : f4zero();
    float4 v100 = (bz1 && by0 && bx0) ? vol[r10 + cx0] : f4zero();
    float4 v101 = (bz1 && by0 && bx1) ? vol[r10 + cx1] : f4zero();
    float4 v110 = (bz1 && by1 && bx0) ? vol[r11 + cx0] : f4zero();
    float4 v111 = (bz1 && by1 && bx1) ? vol[r11 + cx1] : f4zero();

    float4 a0 = lerp4(v000, v001, wx);
    float4 a1 = lerp4(v010, v011, wx);
    float4 a2 = lerp4(v100, v101, wx);
    float4 a3 = lerp4(v110, v111, wx);
    float4 b0 = lerp4(a0, a1, wy);
    float4 b1 = lerp4(a2, a3, wy);
    return lerp4(b0, b1, wz);
}

// ---------------------------------------------------------------------------
// Planar fallback sampler: vol[c*GGG + z*GG + y*G + x], NC channels.
// ---------------------------------------------------------------------------
template <int NC>
__device__ __forceinline__ void tri_sample_p(const float* __restrict__ vol,
                                             float px, float py, float pz,
                                             float out[NC]) {
    const float s = 0.5f * (float)(G - 1);
    float fx = fmaf(px, s, s);
    float fy = fmaf(py, s, s);
    float fz = fmaf(pz, s, s);
    float x0f = floorf(fx), y0f = floorf(fy), z0f = floorf(fz);
    float wx = fx - x0f, wy = fy - y0f, wz = fz - z0f;
    int x0 = (int)x0f, y0 = (int)y0f, z0 = (int)z0f;
    int x1 = x0 + 1,  y1 = y0 + 1,  z1 = z0 + 1;
    bool bx0 = (unsigned)x0 < (unsigned)G, bx1 = (unsigned)x1 < (unsigned)G;
    bool by0 = (unsigned)y0 < (unsigned)G, by1 = (unsigned)y1 < (unsigned)G;
    bool bz0 = (unsigned)z0 < (unsigned)G, bz1 = (unsigned)z1 < (unsigned)G;
    int cx0 = min(max(x0, 0), G - 1), cx1 = min(max(x1, 0), G - 1);
    int cy0 = min(max(y0, 0), G - 1), cy1 = min(max(y1, 0), G - 1);
    int cz0 = min(max(z0, 0), G - 1), cz1 = min(max(z1, 0), G - 1);

    float w000 = (1.f - wz) * (1.f - wy) * (1.f - wx) * ((bz0 && by0 && bx0) ? 1.f : 0.f);
    float w001 = (1.f - wz) * (1.f - wy) * wx         * ((bz0 && by0 && bx1) ? 1.f : 0.f);
    float w010 = (1.f - wz) * wy * (1.f - wx)         * ((bz0 && by1 && bx0) ? 1.f : 0.f);
    float w011 = (1.f - wz) * wy * wx                 * ((bz0 && by1 && bx1) ? 1.f : 0.f);
    float w100 = wz * (1.f - wy) * (1.f - wx)         * ((bz1 && by0 && bx0) ? 1.f : 0.f);
    float w101 = wz * (1.f - wy) * wx                 * ((bz1 && by0 && bx1) ? 1.f : 0.f);
    float w110 = wz * wy * (1.f - wx)                 * ((bz1 && by1 && bx0) ? 1.f : 0.f);
    float w111 = wz * wy * wx                         * ((bz1 && by1 && bx1) ? 1.f : 0.f);

    int i000 = cz0 * GG + cy0 * G + cx0;
    int i001 = cz0 * GG + cy0 * G + cx1;
    int i010 = cz0 * GG + cy1 * G + cx0;
    int i011 = cz0 * GG + cy1 * G + cx1;
    int i100 = cz1 * GG + cy0 * G + cx0;
    int i101 = cz1 * GG + cy0 * G + cx1;
    int i110 = cz1 * GG + cy1 * G + cx0;
    int i111 = cz1 * GG + cy1 * G + cx1;

#pragma unroll
    for (int c = 0; c < NC; ++c) {
        const float* vc = vol + c * GGG;
        float acc = w000 * vc[i000];
        acc = fmaf(w001, vc[i001], acc);
        acc = fmaf(w010, vc[i010], acc);
        acc = fmaf(w011, vc[i011], acc);
        acc = fmaf(w100, vc[i100], acc);
        acc = fmaf(w101, vc[i101], acc);
        acc = fmaf(w110, vc[i110], acc);
        acc = fmaf(w111, vc[i111], acc);
        out[c] = acc;
    }
}

// ---------------------------------------------------------------------------
// One-time planar -> interleaved float4 re-layout (streaming, ~125 MB HBM,
// ~5 us at 23.3 TB/s — DRAM-rate-limited, so the store path choice is
// perf-neutral). Planar reads: touch-once -> TH=NT so they don't evict the
// gather working set. Interleaved writes are drained through LDS with the
// gfx1250 async-store engine (global_store_async_from_lds_b128, ASYNCcnt):
//   ds_store_b128 -> s_wait_dscnt 0 (async engine is unordered w.r.t. DS ops,
//   LDS data must be visible first) -> async store -> s_wait_asynccnt 0.
// GGG = 2^21 is divisible by 256 -> every block is full, EXEC all-ones.
// ---------------------------------------------------------------------------
__global__ void __launch_bounds__(256)
interleave_kernel(const float* __restrict__ warp, const float* __restrict__ tmpl,
                  float4* __restrict__ warp4, float4* __restrict__ tmpl4) {
    __shared__ float4 shw[256];
    __shared__ float4 sht[256];
    const int t = threadIdx.x;
    const int i = blockIdx.x * 256 + t;

    float w0 = __builtin_nontemporal_load(warp + i);
    float w1 = __builtin_nontemporal_load(warp + GGG + i);
    float w2 = __builtin_nontemporal_load(warp + 2 * GGG + i);
    float t0 = __builtin_nontemporal_load(tmpl + i);
    float t1 = __builtin_nontemporal_load(tmpl + GGG + i);
    float t2 = __builtin_nontemporal_load(tmpl + 2 * GGG + i);
    float t3 = __builtin_nontemporal_load(tmpl + 3 * GGG + i);

    shw[t] = make_float4(w0, w1, w2, 0.f);   // ds_store_b128
    sht[t] = make_float4(t0, t1, t2, t3);    // ds_store_b128

    unsigned lds_w = (unsigned)(uintptr_t)&shw[t];
    unsigned lds_t = (unsigned)(uintptr_t)&sht[t];
    unsigned long long g_w = (unsigned long long)(uintptr_t)(warp4 + i);
    unsigned long long g_t = (unsigned long long)(uintptr_t)(tmpl4 + i);

    asm volatile("s_wait_dscnt 0x0" ::: "memory");
    asm volatile("global_store_async_from_lds_b128 %0, %1, off"
                 :: "v"(g_w), "v"(lds_w) : "memory");
    asm volatile("global_store_async_from_lds_b128 %0, %1, off"
                 :: "v"(g_t), "v"(lds_t) : "memory");
    asm volatile("s_wait_asynccnt 0x0" ::: "memory");
}

// ---------------------------------------------------------------------------
// Main raymarch: one lane per ray, 8192 wave32 waves.
// ---------------------------------------------------------------------------
template <bool INTER>
__global__ void __launch_bounds__(256)
raymarch_kernel(const float* __restrict__ raypos, const float* __restrict__ raydir,
                const float* __restrict__ tminmax,
                const void* __restrict__ warpv, const void* __restrict__ tmplv,
                const int* __restrict__ nsteps_p, float* __restrict__ out) {
    int i = blockIdx.x * blockDim.x + threadIdx.x;
    if (i >= NRAYS) return;

    const float dt = DTC;
    // Ray state is touch-once: non-temporal loads keep L2 for the volumes.
    float px = __builtin_nontemporal_load(raypos + 3 * i + 0);
    float py = __builtin_nontemporal_load(raypos + 3 * i + 1);
    float pz = __builtin_nontemporal_load(raypos + 3 * i + 2);
    float dx = __builtin_nontemporal_load(raydir + 3 * i + 0);
    float dy = __builtin_nontemporal_load(raydir + 3 * i + 1);
    float dz = __builtin_nontemporal_load(raydir + 3 * i + 2);
    float tmn = __builtin_nontemporal_load(tminmax + 2 * i);
    float t0 = floorf(tmn / dt) * dt;
    px = fmaf(dx, t0, px);
    py = fmaf(dy, t0, py);
    pz = fmaf(dz, t0, pz);

    const int nsteps = *nsteps_p;
    float r = 0.f, g = 0.f, b = 0.f, a = 0.f;

    const float4* warp4 = (const float4*)warpv;
    const float4* tmpl4 = (const float4*)tmplv;
    const float* warpp = (const float*)warpv;
    const float* tmplp = (const float*)tmplv;

    for (int s = 0; s < nsteps; ++s) {
        bool valid = (px > -1.f) & (px < 1.f) & (py > -1.f) & (py < 1.f) &
                     (pz > -1.f) & (pz < 1.f);

        if constexpr (INTER) {
            // Prefetch all 4 (y,z) corner rows of the NEXT step's warp sample.
            // Each 128 B line covers both x corners (2 x 16 B voxels), so 4
            // global_prefetch_b8 cover all 8 corners one iteration ahead. The
            // un-warped path advances deterministically (~1.3 voxels/step).
            const float sc = 0.5f * (float)(G - 1);
            float nfx = fmaf(fmaf(dx, dt, px), sc, sc);
            float nfy = fmaf(fmaf(dy, dt, py), sc, sc);
            float nfz = fmaf(fmaf(dz, dt, pz), sc, sc);
            int nx0 = min(max((int)nfx, 0), G - 1);
            int ny0 = min(max((int)nfy, 0), G - 1);
            int nz0 = min(max((int)nfz, 0), G - 1);
            int ny1 = min(ny0 + 1, G - 1);
            int nz1 = min(nz0 + 1, G - 1);
            const char* base = (const char*)warp4;
            __builtin_prefetch(base + (((size_t)(nz0 * GG + ny0 * G + nx0)) << 4), 0, 0);
            __builtin_prefetch(base + (((size_t)(nz0 * GG + ny1 * G + nx0)) << 4), 0, 0);
            __builtin_prefetch(base + (((size_t)(nz1 * GG + ny0 * G + nx0)) << 4), 0, 0);
            __builtin_prefetch(base + (((size_t)(nz1 * GG + ny1 * G + nx0)) << 4), 0, 0);
        }

        if (valid && a < 1.0f) {   // alpha==1 => contrib==0 forever (template >= 0)
            float vr, vg, vb, va;
            if constexpr (INTER) {
                float4 sp  = tri_sample_i(warp4, px, py, pz);
                float4 val = tri_sample_i(tmpl4, sp.x, sp.y, sp.z);
                vr = val.x; vg = val.y; vb = val.z; va = val.w;
            } else {
                float sp[3];
                tri_sample_p<3>(warpp, px, py, pz, sp);
                float val[4];
                tri_sample_p<4>(tmplp, sp[0], sp[1], sp[2], val);
                vr = val[0]; vg = val[1]; vb = val[2]; va = val[3];
            }
            float contrib = fminf(fmaf(va, dt, a), 1.f) - a;
            r = fmaf(vr, contrib, r);
            g = fmaf(vg, contrib, g);
            b = fmaf(vb, contrib, b);
            a += contrib;
        }

        px = fmaf(dx, dt, px);
        py = fmaf(dy, dt, py);
        pz = fmaf(dz, dt, pz);

        if (__all(a >= 1.0f)) break;   // whole-wave saturated: exact early exit
    }

    // Output is write-once: stream past L2 (TH=NT store).
    __builtin_nontemporal_store(r, out + 0 * NRAYS + i);
    __builtin_nontemporal_store(g, out + 1 * NRAYS + i);
    __builtin_nontemporal_store(b, out + 2 * NRAYS + i);
    __builtin_nontemporal_store(a, out + 3 * NRAYS + i);
}

extern "C" void kernel_launch(void* const* d_in, const int* in_sizes, int n_in,
                              void* d_out, int out_size, void* d_ws, size_t ws_size,
                              hipStream_t stream) {
    const float* raypos  = (const float*)d_in[0];
    const float* raydir  = (const float*)d_in[1];
    const float* tminmax = (const float*)d_in[2];
    const float* warp    = (const float*)d_in[3];
    const float* tmpl    = (const float*)d_in[4];
    const int*   nsteps  = (const int*)d_in[5];
    float* out = (float*)d_out;

    const size_t need = (size_t)GGG * sizeof(float4) * 2;   // 67 MB: L2-resident
    dim3 blk(256);
    dim3 grd_rays((NRAYS + 255) / 256);

    if (ws_size >= need) {
        float4* warp4 = (float4*)d_ws;
        float4* tmpl4 = warp4 + GGG;
        interleave_kernel<<<GGG / 256, blk, 0, stream>>>(warp, tmpl, warp4, tmpl4);
        raymarch_kernel<true><<<grd_rays, blk, 0, stream>>>(
            raypos, raydir, tminmax, (const void*)warp4, (const void*)tmpl4, nsteps, out);
    } else {
        raymarch_kernel<false><<<grd_rays, blk, 0, stream>>>(
            raypos, raydir, tminmax, (const void*)warp, (const void*)tmpl, nsteps, out);
    }
}